// LocalGainNorm_89902255440493
// MI455X (gfx1250) — compile-verified
//
#include <hip/hip_runtime.h>
#include <cstdint>
#include <cstddef>

// LocalGainNorm: out[b,t,c] = x / (sqrt(s_t + eps) + eps),
// s_t = 0.98 s_{t-1} + 0.02 x^2, causal scan over t per (b,c) series.
// Pure streaming kernel: 512 MiB traffic -> ~23 us floor @ 23.3 TB/s.
// CDNA5 path: 4-deep ring of async global->LDS copies (ASYNCcnt-gated),
// ~16 KB in flight per wave x 1024 waves ~= 16 MB device-wide (latency*BW).

#define EMA_ALPHA 0.98f
#define EMA_BETA  0.02f
#define EMA_EPS   1e-6f

constexpr int Bdim = 64;
constexpr int Tdim = 2048;
constexpr int Cdim = 512;

constexpr int TT     = 32;            // time steps per chunk
constexpr int NCHUNK = Tdim / TT;     // 64
constexpr int NBUF   = 4;             // ring depth (3 chunks prefetched ahead)
constexpr int CPW    = 32;            // channels per wave (== lanes)
constexpr int WPB    = 2;             // waves per block
constexpr int BLOCK  = 32 * WPB;      // 64 threads

__global__ __launch_bounds__(BLOCK)
void LocalGainNorm_kernel(const float* __restrict__ x, float* __restrict__ out) {
    // Per-wave ring of tiles: [wave][buf][tt][lane]  (4 KiB per buffer)
    __shared__ float smem[WPB][NBUF][TT][CPW];

    const int lane = threadIdx.x & 31;
    const int wv   = threadIdx.x >> 5;
    const int wid  = blockIdx.x * WPB + wv;          // 0 .. 1023 (uniform per wave)
    const int b    = wid / (Cdim / CPW);
    const int c0   = (wid % (Cdim / CPW)) * CPW;

    const float* xb = x + (size_t)b * Tdim * Cdim;           // SGPR base (uniform)
    float*       ob = out + (size_t)b * Tdim * Cdim + c0 + lane;

    // 8 B128 segments per chunk per lane: seg = lane + 32*j; t = seg>>3, oct = seg&7.
    // Global byte offset within batch: ((k*TT + t)*C + c0 + 4*oct)*4  (16B aligned).
    // LDS byte offset within buffer:    seg*16  (matches [tt][lane] float layout).
    const uint32_t lds_base = (uint32_t)(uintptr_t)&smem[wv][0][0][0];
    uint32_t goff[8], loff[8];
#pragma unroll
    for (int j = 0; j < 8; ++j) {
        const int seg = lane + 32 * j;
        const int t   = seg >> 3;
        const int oct = seg & 7;
        goff[j] = (uint32_t)((t * Cdim + c0 + 4 * oct) * 4);
        loff[j] = lds_base + (uint32_t)(seg * 16);
    }
    const uint32_t chunk_gbytes = (uint32_t)(TT * Cdim * 4);  // 64 KiB of time per chunk
    const uint32_t buf_bytes    = (uint32_t)(TT * CPW * 4);   // 4 KiB per buffer

    auto issue_chunk = [&](int k) {
        const uint32_t gadd = (uint32_t)k * chunk_gbytes;
        const uint32_t ladd = (uint32_t)(k & (NBUF - 1)) * buf_bytes;
#pragma unroll
        for (int j = 0; j < 8; ++j) {
            uint32_t l = loff[j] + ladd;
            uint32_t g = goff[j] + gadd;
            // vdst = per-lane LDS byte address, vaddr = 32-bit byte offset, saddr = 64-bit base
            asm volatile("global_load_async_to_lds_b128 %0, %1, %2"
                         :: "v"(l), "v"(g), "s"(xb)
                         : "memory");
        }
    };

    // Prologue: 3 chunks in flight (24 outstanding async B128 loads).
    issue_chunk(0);
    issue_chunk(1);
    issue_chunk(2);

    float s = 0.0f;
    for (int k = 0; k < NCHUNK; ++k) {
        const int ahead = NCHUNK - 1 - k;   // chunks still outstanding once k is consumed
        if (k + 3 < NCHUNK) {
            issue_chunk(k + 3);
            // 24 loads of chunks k+1..k+3 outstanding; <=24 => all of chunk k landed.
            asm volatile("s_wait_asynccnt 24" ::: "memory");
        } else if (ahead >= 2) {
            asm volatile("s_wait_asynccnt 16" ::: "memory");
        } else if (ahead == 1) {
            asm volatile("s_wait_asynccnt 8" ::: "memory");
        } else {
            asm volatile("s_wait_asynccnt 0" ::: "memory");
        }

        const float* buf = &smem[wv][k & (NBUF - 1)][0][0];
        float* op = ob + (size_t)k * TT * Cdim;
#pragma unroll 8
        for (int tt = 0; tt < TT; ++tt) {
            float xv = buf[tt * CPW + lane];                 // ds_load_2addr_b32, conflict-free
            s = EMA_ALPHA * s + EMA_BETA * (xv * xv);        // carried recurrence (1 FMA chain)
            float g = __builtin_amdgcn_sqrtf(s + EMA_EPS);   // v_sqrt_f32 (TRANS, off chain)
            float r = __builtin_amdgcn_rcpf(g + EMA_EPS);    // v_rcp_f32  (TRANS, off chain)
            __builtin_nontemporal_store(xv * r, op + (size_t)tt * Cdim);
        }
    }
}

extern "C" void kernel_launch(void* const* d_in, const int* in_sizes, int n_in,
                              void* d_out, int out_size, void* d_ws, size_t ws_size,
                              hipStream_t stream) {
    (void)in_sizes; (void)n_in; (void)d_ws; (void)ws_size; (void)out_size;
    const float* x = (const float*)d_in[0];
    float* out = (float*)d_out;
    const int total_waves = Bdim * (Cdim / CPW);     // 1024
    const int grid = total_waves / WPB;              // 512 blocks x 64 threads
    LocalGainNorm_kernel<<<grid, BLOCK, 0, stream>>>(x, out);
}